// LearnablePCEN_36051955482627
// MI455X (gfx1250) — compile-verified
//
#include <hip/hip_runtime.h>
#include <math.h>

// ---------------------------------------------------------------------------
// PCEN on MI455X (gfx1250, wave32).
//
//   m_t = (1-s) m_{t-1} + s x_t   (m_0 = x_0), then
//   pcen = (x/(EPS+m)^a + d)^r - d^r
//
// 16-step IIR chunk == 16x16 lower-triangular Toeplitz matmul, computed with
// V_WMMA_F32_16X16X4_F32 (4 K-groups, exact f32 accumulate). Chunks chain with
// one FMA per 16 steps; T splits into 50 segments of 160 whose exact carries
// come from a tiny 3-phase scan. B-fragment loads are double-buffered across
// chunks so WMMA overlaps the next chunk's global loads.
// ---------------------------------------------------------------------------

#define EPSV    1e-6f
#define B_      64
#define C_      80
#define T_      8000
#define SEG_LEN 160
#define NSEG    50      // 50 * 160 = 8000
#define NCHUNK  10      // 10 * 16  = 160
#define BG      4       // 4 batch-groups of 16 (B = 64)
#define WPB     4       // waves per block

typedef __attribute__((ext_vector_type(2))) float v2f;
typedef __attribute__((ext_vector_type(8))) float v8f;

__device__ __forceinline__ float sigm(float z) { return 1.0f / (1.0f + expf(-z)); }

// A-matrix 16x4 f32 layout: lane L -> row M = L&15; K = 2*(L>>4) + v.
__device__ __forceinline__ void build_afrags(float s, float q, int t, int half, v2f a[4]) {
#pragma unroll
    for (int j = 0; j < 4; ++j) {
#pragma unroll
        for (int v = 0; v < 2; ++v) {
            int k  = 4 * j + 2 * half + v;
            int dd = t - k;
            a[j][v] = (dd >= 0) ? s * powf(q, (float)dd) : 0.0f;
        }
    }
}

// Issue the 4 raw B-fragment b64 loads for one chunk (no relu yet, so the
// loads can stay in flight while previous chunk's WMMAs run).
// B-matrix 4x16 layout: lane L -> col N = L&15; K = v + 2*(L>>4).
__device__ __forceinline__ void load_bfrags(const float* __restrict__ p0, float2 bf[4]) {
#pragma unroll
    for (int j = 0; j < 4; ++j)
        bf[j] = *(const float2*)(p0 + 4 * j);
}

// 4 chained WMMAs over one chunk; ReLU fused at operand use.
__device__ __forceinline__ v8f chunk_wmma(const v2f a[4], const float2 bf[4]) {
    v8f acc = {0.f, 0.f, 0.f, 0.f, 0.f, 0.f, 0.f, 0.f};
#pragma unroll
    for (int j = 0; j < 4; ++j) {
        v2f b;
        b[0] = fmaxf(bf[j].x, 0.0f);
        b[1] = fmaxf(bf[j].y, 0.0f);
        acc = __builtin_amdgcn_wmma_f32_16x16x4_f32(false, a[j], false, b,
                                                    (short)0, acc, false, false);
    }
    return acc;
}

// --------------------------- Kernel 1: segment summaries --------------------
__global__ __launch_bounds__(32 * WPB)
void pcen_seg_summary(const float* __restrict__ f, const float* __restrict__ logit_s,
                      float* __restrict__ mlf) {
    const int wid  = blockIdx.x * WPB + (threadIdx.x >> 5);
    const int bg   = wid % BG;
    const int seg  = (wid / BG) % NSEG;
    const int c    = wid / (BG * NSEG);
    const int lane = threadIdx.x & 31;
    const int n    = lane & 15;
    const int half = lane >> 4;

    const float s = sigm(logit_s[c]);
    const float q = 1.0f - s;
    v2f a[4];
    build_afrags(s, q, n, half, a);
    const float dec16 = powf(q, 16.0f);

    const int bIdx = bg * 16 + n;
    const float* xrow = f + ((size_t)bIdx * C_ + c) * T_ + (size_t)seg * SEG_LEN + 2 * half;

    float2 cur[4], nxt[4];
    load_bfrags(xrow, cur);

    float LV = 0.0f;                                   // zero-carry local value
    for (int g = 0; g < NCHUNK; ++g) {
        if (g + 1 < NCHUNK)
            load_bfrags(xrow + (g + 1) * 16, nxt);     // overlap with WMMAs below
        v8f acc = chunk_wmma(a, cur);
        float m15 = __shfl(acc[7], 16 + n, 32);        // m_local[15]: hi lanes, r=7
        LV = fmaf(dec16, LV, m15);
#pragma unroll
        for (int j = 0; j < 4; ++j) cur[j] = nxt[j];
    }
    if (lane < 16)
        mlf[((size_t)c * NSEG + seg) * B_ + bIdx] = LV;
}

// --------------------------- Kernel 2: carry propagation --------------------
__global__ void pcen_carry(const float* __restrict__ f, const float* __restrict__ logit_s,
                           const float* __restrict__ mlf, float* __restrict__ cin) {
    int tid = blockIdx.x * blockDim.x + threadIdx.x;
    if (tid >= B_ * C_) return;
    int b = tid / C_;
    int c = tid % C_;
    const float q    = 1.0f - sigm(logit_s[c]);
    const float decL = powf(q, (float)SEG_LEN);
    // carry_in(0) = x_0 makes the uniform-weight scan satisfy m_0 = x_0 exactly.
    float carry = fmaxf(f[((size_t)b * C_ + c) * T_], 0.0f);
    cin[(size_t)c * NSEG * B_ + b] = carry;
    for (int g = 1; g < NSEG; ++g) {
        carry = fmaf(decL, carry, mlf[((size_t)c * NSEG + g - 1) * B_ + b]);
        cin[((size_t)c * NSEG + g) * B_ + b] = carry;
    }
}

// --------------------------- Kernel 3: fix-up + PCEN ------------------------
__global__ __launch_bounds__(32 * WPB)
void pcen_apply(const float* __restrict__ f, const float* __restrict__ logit_s,
                const float* __restrict__ alpha, const float* __restrict__ delta,
                const float* __restrict__ log_r, const float* __restrict__ cin,
                float* __restrict__ out) {
    const int wid  = blockIdx.x * WPB + (threadIdx.x >> 5);
    const int bg   = wid % BG;
    const int seg  = (wid / BG) % NSEG;
    const int c    = wid / (BG * NSEG);
    const int lane = threadIdx.x & 31;
    const int n    = lane & 15;
    const int half = lane >> 4;

    const float s    = sigm(logit_s[c]);
    const float q    = 1.0f - s;
    const float aexp = fminf(fmaxf(alpha[c], 0.1f), 1.0f);
    const float dpar = fmaxf(delta[c], 0.1f);
    const float rpar = fminf(fmaxf(expf(log_r[c]), 0.05f), 1.5f);
    const float drp  = powf(dpar, rpar);

    v2f a[4];
    build_afrags(s, q, n, half, a);
    float qp[8];                                       // q^(t_local+1), D row M = r + 8*half
#pragma unroll
    for (int r = 0; r < 8; ++r) qp[r] = powf(q, (float)(r + 8 * half + 1));

    const int bIdx = bg * 16 + n;
    const float* xrow = f   + ((size_t)bIdx * C_ + c) * T_ + (size_t)seg * SEG_LEN;
    float*       orow = out + ((size_t)bIdx * C_ + c) * T_ + (size_t)seg * SEG_LEN;
    const float* xb   = xrow + 2 * half;               // B-fragment base

    float cc = cin[((size_t)c * NSEG + seg) * B_ + bIdx];

    float2 cur[4], nxt[4];
    load_bfrags(xb, cur);

    for (int g = 0; g < NCHUNK; ++g) {
        // Independent loads first, so they fly behind the WMMA chain:
        const float4* xp = (const float4*)(xrow + g * 16 + 8 * half);
        float4 x0 = xp[0], x1 = xp[1];                 // this lane's 8 output x's
        if (g + 1 < NCHUNK)
            load_bfrags(xb + (g + 1) * 16, nxt);       // next chunk's B-fragments

        v8f acc = chunk_wmma(a, cur);

        float xs[8] = {x0.x, x0.y, x0.z, x0.w, x1.x, x1.y, x1.z, x1.w};
        float res[8];
        float m7 = 0.0f;
#pragma unroll
        for (int r = 0; r < 8; ++r) {
            float m = fmaf(qp[r], cc, acc[r]);         // m = m_local + q^(t+1)*carry
            if (r == 7) m7 = m;
            float xv   = fmaxf(xs[r], 0.0f);
            float base = xv * powf(EPSV + m, -aexp) + dpar;
            res[r] = powf(base, rpar) - drp;
        }
        cc = __shfl(m7, 16 + n, 32);                   // m[15] from hi lanes

        float4* op = (float4*)(orow + g * 16 + 8 * half);
        op[0] = make_float4(res[0], res[1], res[2], res[3]);
        op[1] = make_float4(res[4], res[5], res[6], res[7]);
#pragma unroll
        for (int j = 0; j < 4; ++j) cur[j] = nxt[j];
    }
}

// ---------------------------------------------------------------------------
extern "C" void kernel_launch(void* const* d_in, const int* in_sizes, int n_in,
                              void* d_out, int out_size, void* d_ws, size_t ws_size,
                              hipStream_t stream) {
    const float* f       = (const float*)d_in[0];
    const float* logit_s = (const float*)d_in[1];
    const float* alpha   = (const float*)d_in[2];
    const float* delta   = (const float*)d_in[3];
    const float* log_r   = (const float*)d_in[4];
    float*       out     = (float*)d_out;

    float* mlf = (float*)d_ws;                       // [C, NSEG, B]
    float* cin = mlf + (size_t)C_ * NSEG * B_;       // [C, NSEG, B]   (~2 MB total)

    const int nwaves  = C_ * NSEG * BG;              // 16,000 independent waves
    const int nblocks = nwaves / WPB;                // 4,000 blocks of 128 threads

    pcen_seg_summary<<<nblocks, 32 * WPB, 0, stream>>>(f, logit_s, mlf);
    pcen_carry<<<(B_ * C_ + 255) / 256, 256, 0, stream>>>(f, logit_s, mlf, cin);
    pcen_apply<<<nblocks, 32 * WPB, 0, stream>>>(f, logit_s, alpha, delta, log_r, cin, out);
}